// MultiBehavioralGNN_72112500899858
// MI455X (gfx1250) — compile-verified
//
#include <hip/hip_runtime.h>

// ---------------------------------------------------------------------------
// MultiBehavioralGNN for MI455X (gfx1250, wave32, WMMA)
// Sizes fixed by the reference:
#define N1 50000
#define N_LAYERS 3
#define N2 (N1 * N_LAYERS)   // 150000
#define E1 800000
#define E2 2400000
#define D 64
#define LEAKY 0.01f

typedef __attribute__((ext_vector_type(2))) float v2f;
typedef __attribute__((ext_vector_type(8))) float v8f;

// ---------------------------------------------------------------------------
// Zero a float4 region of workspace.
__global__ void zero_f4_kernel(float4* __restrict__ p, int n4) {
    int i = blockIdx.x * blockDim.x + threadIdx.x;
    if (i < n4) p[i] = make_float4(0.f, 0.f, 0.f, 0.f);
}

// Degree accumulation: one thread per edge, atomic +1 on out/in degree.
__global__ void degree_kernel(const int* __restrict__ src, const int* __restrict__ dst,
                              int nedges, float* __restrict__ deg_out,
                              float* __restrict__ deg_in) {
    int e = blockIdx.x * blockDim.x + threadIdx.x;
    if (e >= nedges) return;
    atomicAdd(&deg_out[src[e]], 1.0f);
    atomicAdd(&deg_in[dst[e]], 1.0f);
}

// In-place deg -> rsqrt(max(deg, 1))
__global__ void rsqrt_kernel(float* __restrict__ d, int n) {
    int i = blockIdx.x * blockDim.x + threadIdx.x;
    if (i >= n) return;
    float v = d[i];
    v = v < 1.0f ? 1.0f : v;
    d[i] = rsqrtf(v);
}

// Edge scatter: 16 threads per edge, float4 gather of source features scaled by
// norm_src, 4 f32 atomic adds into agg[dst].  n1mod > 0 applies the kron-tile
// row mapping (feature row = src % n1mod, src in [0, 3*n1mod)).
__global__ void scatter_kernel(const float* __restrict__ feat,
                               const int* __restrict__ src, const int* __restrict__ dst,
                               int nedges, const float* __restrict__ norm_src,
                               float* __restrict__ agg, int n1mod) {
    int tid = blockIdx.x * blockDim.x + threadIdx.x;
    int e = tid >> 4;
    if (e >= nedges) return;
    int q = tid & 15;
    int s = src[e];
    int t = dst[e];
    float w = norm_src[s];
    int fs = s;
    if (n1mod > 0) {                 // cheap mod for s in [0, 3*n1mod)
        if (fs >= 2 * n1mod) fs -= 2 * n1mod;
        else if (fs >= n1mod) fs -= n1mod;
    }
    float4 v = ((const float4*)(feat + (size_t)fs * D))[q];
    float* ao = agg + (size_t)t * D + q * 4;
    atomicAdd(ao + 0, v.x * w);
    atomicAdd(ao + 1, v.y * w);
    atomicAdd(ao + 2, v.z * w);
    atomicAdd(ao + 3, v.w * w);
}

// ---------------------------------------------------------------------------
// GEMM epilogue:  out[r,:] = leaky( (agg[r,:] * norm_in[r]) @ W + b )
// One wave computes a 16x64 output tile via V_WMMA_F32_16X16X4_F32:
// 4 N-tiles x 16 K-steps (K=4) = 64 WMMA per wave.
// W is staged in LDS k-pair interleaved: sWp[k/2][n] = {W[k][n], W[k+1][n]}
// so every B fragment is ONE aligned ds_load_b64 straight into the operand
// VGPR pair (no register shuffling between DS load and WMMA).
__global__ __launch_bounds__(128) void gemm_wmma_kernel(
        const float* __restrict__ agg, const float* __restrict__ norm_in,
        const float* __restrict__ W, const float* __restrict__ bias,
        float* __restrict__ out, int ntiles) {
    __shared__ v2f   sWp[(D / 2) * D];   // 32 k-pairs x 64 cols = 16 KB
    __shared__ float sb[D];

    const int tid = threadIdx.x;
    // Cooperative interleaved stage of W: 512 tasks (32 k-pairs x 16 col-quads),
    // 4 per thread.  Each task: two row-major float4 loads, two 16B LDS stores.
    {
        const float4* Wv = (const float4*)W;   // Wv[k*16 + n4] = W[k][4n4..4n4+3]
#pragma unroll
        for (int i = 0; i < 4; ++i) {
            int task = tid + i * 128;
            int k2 = task >> 4;               // k-pair index, 0..31
            int n4 = task & 15;               // column quad, 0..15
            float4 w0 = Wv[(2 * k2 + 0) * 16 + n4];
            float4 w1 = Wv[(2 * k2 + 1) * 16 + n4];
            v2f* dsti = &sWp[k2 * D + n4 * 4];
            dsti[0] = (v2f){w0.x, w1.x};
            dsti[1] = (v2f){w0.y, w1.y};
            dsti[2] = (v2f){w0.z, w1.z};
            dsti[3] = (v2f){w0.w, w1.w};
        }
        if (tid < D) sb[tid] = bias[tid];
    }
    __syncthreads();

    const int wave = tid >> 5;
    const int lane = tid & 31;
    const int half = lane >> 4;       // 0: K lo pair, 1: K hi pair (A/B layout)
    const int r    = lane & 15;       // row within tile for A; col within N-tile for B/D
    const int tile = blockIdx.x * 4 + wave;
    if (tile >= ntiles) return;       // wave-uniform: EXEC all-1s inside

    const int row = tile * 16 + r;
    const float nd = norm_in[row];
    const float* __restrict__ arow = agg + (size_t)row * D;

    v8f acc[4] = {};                  // 4 N-tiles of 16x16 f32 accumulators
#pragma unroll
    for (int kk = 0; kk < 16; ++kk) {
        const int kA = kk * 4 + half * 2;          // this lane's two K indices
        v2f a;
        a.x = arow[kA + 0] * nd;                   // A[r, kA],  16x4 f32 layout
        a.y = arow[kA + 1] * nd;                   // A[r, kA+1]
        const v2f* __restrict__ brow = &sWp[(kk * 2 + half) * D + r];
#pragma unroll
        for (int n = 0; n < 4; ++n) {
            v2f bfr = brow[n * 16];                // {W[kA][col], W[kA+1][col]}
            acc[n] = __builtin_amdgcn_wmma_f32_16x16x4_f32(
                false, a, false, bfr, (short)0, acc[n], false, false);
        }
    }

    // C/D layout: VGPR v -> M = v + 8*half, lane -> N = r (within N-tile).
#pragma unroll
    for (int n = 0; n < 4; ++n) {
        const int col = n * 16 + r;
        const float bc = sb[col];
#pragma unroll
        for (int v = 0; v < 8; ++v) {
            const int orow = tile * 16 + half * 8 + v;
            float val = acc[n][v] + bc;
            val = val >= 0.f ? val : val * LEAKY;
            out[(size_t)orow * D + col] = val;
        }
    }
}

// ---------------------------------------------------------------------------
extern "C" void kernel_launch(void* const* d_in, const int* in_sizes, int n_in,
                              void* d_out, int out_size, void* d_ws, size_t ws_size,
                              hipStream_t stream) {
    (void)in_sizes; (void)n_in; (void)out_size; (void)ws_size;
    const float* x    = (const float*)d_in[0];
    const float* WQ   = (const float*)d_in[1];
    const float* bQ   = (const float*)d_in[2];
    const float* WM   = (const float*)d_in[3];
    const float* bM   = (const float*)d_in[4];
    const int*   src1 = (const int*)d_in[5];
    const int*   dst1 = (const int*)d_in[6];
    const int*   src2 = (const int*)d_in[7];
    const int*   dst2 = (const int*)d_in[8];
    float* out = (float*)d_out;

    // Workspace layout (floats)
    float* p         = (float*)d_ws;
    float* norm_out1 = p;                          // N1
    float* norm_in1  = norm_out1 + N1;             // N1
    float* norm_out2 = norm_in1 + N1;              // N2
    float* norm_in2  = norm_out2 + N2;             // N2
    float* agg1      = norm_in2 + N2;              // N1*D
    float* agg2      = agg1 + (size_t)N1 * D;      // N2*D
    float* h1        = agg2 + (size_t)N2 * D;      // N1*D

    const int nzero   = 2 * (N1 + N2) + N1 * D + N2 * D;   // degrees + both agg
    const int nzero4  = nzero / 4;                          // divisible by 4
    const int nnorm   = 2 * (N1 + N2);

    // 1) zero degree + aggregation buffers
    zero_f4_kernel<<<(nzero4 + 255) / 256, 256, 0, stream>>>((float4*)p, nzero4);

    // 2) degrees for both graphs
    degree_kernel<<<(E1 + 255) / 256, 256, 0, stream>>>(src1, dst1, E1, norm_out1, norm_in1);
    degree_kernel<<<(E2 + 255) / 256, 256, 0, stream>>>(src2, dst2, E2, norm_out2, norm_in2);

    // 3) deg -> rsqrt(max(deg,1)) in place
    rsqrt_kernel<<<(nnorm + 255) / 256, 256, 0, stream>>>(p, nnorm);

    // 4) layer 1 scatter: agg1[dst] += x[src] * norm_out1[src]
    {
        const int nthr = E1 * 16;
        scatter_kernel<<<(nthr + 255) / 256, 256, 0, stream>>>(
            x, src1, dst1, E1, norm_out1, agg1, 0);
    }

    // 5) layer 1 GEMM + bias + leaky -> h1
    {
        const int ntiles = N1 / 16;   // 3125
        gemm_wmma_kernel<<<(ntiles + 3) / 4, 128, 0, stream>>>(
            agg1, norm_in1, WQ, bQ, h1, ntiles);
    }

    // 6) layer 2 scatter with kron tiling: agg2[dst] += h1[src % N1] * norm_out2[src]
    {
        const int nthr = E2 * 16;
        scatter_kernel<<<(nthr + 255) / 256, 256, 0, stream>>>(
            h1, src2, dst2, E2, norm_out2, agg2, N1);
    }

    // 7) layer 2 GEMM + bias + leaky -> out
    {
        const int ntiles = N2 / 16;   // 9375
        gemm_wmma_kernel<<<(ntiles + 3) / 4, 128, 0, stream>>>(
            agg2, norm_in2, WM, bM, out, ntiles);
    }
}